// ScaledDotProductAttention_4853313044980
// MI455X (gfx1250) — compile-verified
//
#include <hip/hip_runtime.h>

// Scaled dot-product attention for MI455X (gfx1250), fp32, returning
// (output[B,H,S,D], attn[B,H,S,S]) concatenated in d_out.
// Uses V_WMMA_F32_16X16X4_F32 (wave32 CDNA5 matrix path) for both GEMMs.

typedef float v2f __attribute__((ext_vector_type(2)));
typedef float v8f __attribute__((ext_vector_type(8)));

constexpr int Bb = 4;
constexpr int Hh = 16;
constexpr int Ss = 2048;
constexpr int Dd = 64;
constexpr float INV_TEMP = 0.125f;          // 1 / sqrt(64)
constexpr int SM_STRIDE = Ss + 4;           // pad LDS rows to avoid bank conflicts

__global__ __launch_bounds__(128)
void sdpa_wmma_f32_kernel(const float* __restrict__ q,
                          const float* __restrict__ k,
                          const float* __restrict__ v,
                          const unsigned char* __restrict__ mask,
                          float* __restrict__ out,   // [B,H,S,D]
                          float* __restrict__ attn)  // [B,H,S,S]
{
    extern __shared__ float smem[];              // [16][SM_STRIDE] ~128 KB

    const int lane = threadIdx.x & 31;
    const int wv   = threadIdx.x >> 5;           // wave id 0..3
    const int ln   = lane & 15;                  // N / M index within tile
    const int half = lane >> 4;                  // 0: lanes 0-15, 1: lanes 16-31

    const int qt = blockIdx.x & 127;             // query tile within head (S/16)
    const int bh = blockIdx.x >> 7;              // flattened (b,h)

    const float* Qh = q + (size_t)bh * Ss * Dd;
    const float* Kh = k + (size_t)bh * Ss * Dd;
    const float* Vh = v + (size_t)bh * Ss * Dd;
    const unsigned char* Mh = mask + (size_t)bh * Ss;
    float* Oh = out  + (size_t)bh * Ss * Dd;
    float* Ah = attn + (size_t)bh * Ss * Ss;

    // ---- Q fragments: A-matrix (16x4 f32) layout, K-dim 64 -> 16 fragments.
    // Lane L (L<16): M=L, holds (K=4s, 4s+1); lanes 16-31 hold (K=4s+2, 4s+3).
    const int qrow = qt * 16 + ln;
    v2f qf[16];
#pragma unroll
    for (int s = 0; s < 16; ++s)
        qf[s] = *(const v2f*)(Qh + (size_t)qrow * Dd + 4 * s + 2 * half);

    // ---- Phase 1: scores = mask ? -inf : (Q K^T) * (1/8) -> LDS
    for (int kt = wv; kt < Ss / 16; kt += 4) {
        v8f acc = {};
        // B-matrix (4x16) = K^T: lane n holds K[kbase+n][4s + 2*half .. +1]
        const float* Kb = Kh + (size_t)(kt * 16 + ln) * Dd + 2 * half;
#pragma unroll
        for (int s = 0; s < 16; ++s) {
            v2f bf = *(const v2f*)(Kb + 4 * s);
            acc = __builtin_amdgcn_wmma_f32_16x16x4_f32(
                false, qf[s], false, bf, (short)0, acc, false, false);
        }
        const bool mk = Mh[kt * 16 + ln] != 0;   // mask==True -> -inf
#pragma unroll
        for (int r = 0; r < 8; ++r) {
            // C layout: VGPR r -> row (r + 8*half), col = ln
            float val = mk ? -__builtin_inff() : acc[r] * INV_TEMP;
            smem[(r + half * 8) * SM_STRIDE + kt * 16 + ln] = val;
        }
    }
    __syncthreads();

    // ---- Phase 2: row-wise softmax in LDS, write attn to global (once)
    for (int rr = 0; rr < 4; ++rr) {
        const int r = wv * 4 + rr;
        float* rowp = smem + r * SM_STRIDE;
        float mx = -__builtin_inff();
        for (int i = lane; i < Ss; i += 32)
            mx = fmaxf(mx, rowp[i]);
#pragma unroll
        for (int m = 16; m >= 1; m >>= 1)
            mx = fmaxf(mx, __shfl_xor(mx, m, 32));
        float sum = 0.f;
        for (int i = lane; i < Ss; i += 32) {
            float e = __expf(rowp[i] - mx);      // exp(-inf) = 0 for masked cols
            rowp[i] = e;
            sum += e;
        }
#pragma unroll
        for (int m = 16; m >= 1; m >>= 1)
            sum += __shfl_xor(sum, m, 32);
        const float inv = 1.0f / sum;
        float* arow = Ah + (size_t)(qt * 16 + r) * Ss;
        for (int i = lane; i < Ss; i += 32) {
            float p = rowp[i] * inv;
            rowp[i] = p;                          // keep probs in LDS for P@V
            arow[i] = p;                          // coalesced global write
        }
    }
    __syncthreads();

    // ---- Phase 3: out = P @ V ; each wave owns a 16-wide slice of D
    const int nb = wv * 16;
    v8f oacc = {};
    for (int kt = 0; kt < Ss / 16; ++kt) {
#pragma unroll
        for (int s = 0; s < 4; ++s) {
            const int kk = kt * 16 + s * 4;
            // A fragment (P) from LDS: row ln, (K=kk+2*half, +1)
            v2f af = *(const v2f*)(smem + ln * SM_STRIDE + kk + 2 * half);
            // B fragment (V): VGPR0 = rows {0,2}, VGPR1 = rows {1,3} of the 4xK step
            const float* Vb = Vh + (size_t)(kk + 2 * half) * Dd + nb + ln;
            v2f bf;
            bf[0] = Vb[0];
            bf[1] = Vb[Dd];
            oacc = __builtin_amdgcn_wmma_f32_16x16x4_f32(
                false, af, false, bf, (short)0, oacc, false, false);
        }
    }
#pragma unroll
    for (int r = 0; r < 8; ++r)
        Oh[(size_t)(qt * 16 + r + half * 8) * Dd + nb + ln] = oacc[r];
}

extern "C" void kernel_launch(void* const* d_in, const int* in_sizes, int n_in,
                              void* d_out, int out_size, void* d_ws, size_t ws_size,
                              hipStream_t stream) {
    (void)in_sizes; (void)n_in; (void)out_size; (void)d_ws; (void)ws_size;
    const float* q = (const float*)d_in[0];
    const float* k = (const float*)d_in[1];
    const float* v = (const float*)d_in[2];
    const unsigned char* m = (const unsigned char*)d_in[3];  // bool mask [B,H,1,S]

    float* out  = (float*)d_out;                              // [B,H,S,D]
    float* attn = out + (size_t)Bb * Hh * Ss * Dd;            // [B,H,S,S]

    dim3 grid(Bb * Hh * (Ss / 16));   // 8192 workgroups
    dim3 block(128);                  // 4 wave32s
    size_t shmem = (size_t)16 * SM_STRIDE * sizeof(float);    // ~128 KB of 320 KB/WGP

    hipLaunchKernelGGL(sdpa_wmma_f32_kernel, grid, block, shmem, stream,
                       q, k, v, m, out, attn);
}